// GAT_48206712930429
// MI455X (gfx1250) — compile-verified
//
#include <hip/hip_runtime.h>
#include <hip/hip_bf16.h>
#include <math.h>

#define NNODES 50000
#define NEDGES 800000
#define GAT_SLOPE 0.2f

typedef __attribute__((ext_vector_type(2))) float v2f;
typedef __attribute__((ext_vector_type(8))) float v8f;

// ---- order-preserving float<->uint key for atomicMax-based segment max ----
__device__ __forceinline__ unsigned fkey(float f) {
    unsigned u = __float_as_uint(f);
    return (u & 0x80000000u) ? ~u : (u | 0x80000000u);
}
__device__ __forceinline__ float funkey(unsigned k) {
    return (k & 0x80000000u) ? __uint_as_float(k & 0x7FFFFFFFu)
                             : __uint_as_float(~k);
}

// ---------------------------------------------------------------------------
// WMMA f32 GEMM: C[M,Ncols] = A[M,K] * B[K,Ncols], row-major, fp32.
// One wave computes CT adjacent 16x16 tiles (one row tile x CT col tiles) via
// V_WMMA_F32_16X16X4_F32, stepping K by 4. The (scattered) A fragment is
// loaded once per k-step and reused across all CT column tiles; all CT
// B fragments are loaded into distinct registers BEFORE the wmma chain so the
// backend can clause the loads and drain them with decreasing s_wait_loadcnt
// while wmmas overlap outstanding loads (instead of loadcnt==0 per wmma).
// A-frag (16x4, ISA 7.12.2): lane<16 -> M=lane,    K = {0,1} in v0,v1
//                            lane>=16 -> M=lane-16, K = {2,3}
// B-frag (4x16): v0 lanes0-15 = row K=0 (N across lanes), lanes16-31 = row K=2
//                v1 lanes0-15 = row K=1,                  lanes16-31 = row K=3
// C/D (16x16 f32): VGPR r: lanes0-15 M=r, N=lane; lanes16-31 M=r+8, N=lane-16
// ---------------------------------------------------------------------------
template <int WAVES, int CT>
__global__ __launch_bounds__(32 * WAVES) void gemm_wmma_f32(
    const float* __restrict__ A, const float* __restrict__ B,
    float* __restrict__ C, int M, int K, int Ncols) {
    const int lane = threadIdx.x & 31;
    const int wv   = threadIdx.x >> 5;
    const int hf   = lane >> 4;   // 0 or 1 (half of wave)
    const int l    = lane & 15;
    const int row0 = blockIdx.y * 16;
    const int col0 = (blockIdx.x * WAVES + wv) * (16 * CT);
    if (row0 >= M || col0 >= Ncols) return;  // uniform per wave -> EXEC intact

    v8f acc[CT] = {};
    const float* arow  = A + (size_t)(row0 + l) * K + 2 * hf;
    const float* bbase = B + (size_t)(2 * hf) * Ncols + col0 + l;
    for (int k = 0; k < K; k += 4) {
        // issue all loads for this k-step first (distinct dest registers)
        v2f a = *(const v2f*)(arow + k);  // A[row][k+2h], A[row][k+2h+1]
        v2f b[CT];
#pragma unroll
        for (int t = 0; t < CT; ++t) {
            b[t].x = bbase[(size_t)k * Ncols + 16 * t];          // B[k+2h  ]
            b[t].y = bbase[(size_t)k * Ncols + Ncols + 16 * t];  // B[k+2h+1]
        }
        // then consume: wmmas overlap the still-outstanding later loads
#pragma unroll
        for (int t = 0; t < CT; ++t)
            acc[t] = __builtin_amdgcn_wmma_f32_16x16x4_f32(
                false, a, false, b[t], (short)0, acc[t], false, false);
    }
#pragma unroll
    for (int t = 0; t < CT; ++t) {
        float* crow = C + (size_t)(row0 + 8 * hf) * Ncols + col0 + 16 * t + l;
#pragma unroll
        for (int r = 0; r < 8; ++r) crow[(size_t)r * Ncols] = acc[t][r];
    }
}

// ---------------------------------------------------------------------------
// Per-node attention scalars, wave-per-(node,head) with shuffle reduction:
// as[n,h] = <h[n,h,:], a_src[h,:]>, ad likewise. Coalesced row reads.
// ---------------------------------------------------------------------------
template <int HEADS, int C>
__global__ void node_alpha(const float* __restrict__ h,
                           const float* __restrict__ a_src,
                           const float* __restrict__ a_dst,
                           float* __restrict__ as_, float* __restrict__ ad_,
                           int n) {
    int gwave = (blockIdx.x * blockDim.x + threadIdx.x) >> 5;
    int lane  = threadIdx.x & 31;
    if (gwave >= n * HEADS) return;
    int node = gwave / HEADS, hh = gwave % HEADS;
    const float* row = h + (size_t)node * (HEADS * C) + hh * C;
    const float* vs = a_src + hh * C;
    const float* vd = a_dst + hh * C;
    float ss = 0.f, sd = 0.f;
#pragma unroll
    for (int c = lane; c < C; c += 32) {
        float v = row[c];
        ss += v * vs[c];
        sd += v * vd[c];
    }
#pragma unroll
    for (int off = 16; off > 0; off >>= 1) {   // wave32 reduction
        ss += __shfl_xor(ss, off, 32);
        sd += __shfl_xor(sd, off, 32);
    }
    if (lane == 0) { as_[gwave] = ss; ad_[gwave] = sd; }
}

__device__ __forceinline__ void edge_ids(const int* __restrict__ ei, int nE,
                                         int e, int& s, int& d) {
    if (e < nE) { s = ei[e]; d = ei[nE + e]; }
    else        { s = e - nE; d = e - nE; }   // appended self-loops
}

template <int HEADS>
__device__ __forceinline__ float edge_logit(const float* __restrict__ as_,
                                            const float* __restrict__ ad_,
                                            int s, int d, int hh) {
    float lg = as_[s * HEADS + hh] + ad_[d * HEADS + hh];
    return lg > 0.f ? lg : GAT_SLOPE * lg;   // leaky_relu
}

// pass 1: per-destination max (via ordered-uint atomicMax)
template <int HEADS>
__global__ void edge_max(const int* __restrict__ ei, int nE, int nTot,
                         const float* __restrict__ as_,
                         const float* __restrict__ ad_,
                         unsigned* __restrict__ mkey) {
    int idx = blockIdx.x * blockDim.x + threadIdx.x;
    if (idx >= nTot * HEADS) return;
    int e = idx / HEADS, hh = idx % HEADS, s, d;
    edge_ids(ei, nE, e, s, d);
    atomicMax(&mkey[d * HEADS + hh], fkey(edge_logit<HEADS>(as_, ad_, s, d, hh)));
}

// pass 2: per-destination sum of exp(logit - max)
template <int HEADS>
__global__ void edge_sum(const int* __restrict__ ei, int nE, int nTot,
                         const float* __restrict__ as_,
                         const float* __restrict__ ad_,
                         const unsigned* __restrict__ mkey,
                         float* __restrict__ ssum) {
    int idx = blockIdx.x * blockDim.x + threadIdx.x;
    if (idx >= nTot * HEADS) return;
    int e = idx / HEADS, hh = idx % HEADS, s, d;
    edge_ids(ei, nE, e, s, d);
    float lg = edge_logit<HEADS>(as_, ad_, s, d, hh);
    float m = funkey(mkey[d * HEADS + hh]);
    atomicAdd(&ssum[d * HEADS + hh], __expf(lg - m));
}

// pass 3: out[dst] += alpha * h[src]; ONE WAVE per (edge, head) so that the
// 32 lanes cover consecutive channels -> coalesced gather + clustered atomics.
template <int HEADS, int C>
__global__ void edge_scatter(const int* __restrict__ ei, int nE, int nTot,
                             const float* __restrict__ h,
                             const float* __restrict__ as_,
                             const float* __restrict__ ad_,
                             const unsigned* __restrict__ mkey,
                             const float* __restrict__ ssum,
                             float* __restrict__ out) {
    int gwave = (blockIdx.x * blockDim.x + threadIdx.x) >> 5;
    int lane  = threadIdx.x & 31;
    if (gwave >= nTot * HEADS) return;
    int e = gwave / HEADS, hh = gwave % HEADS, s, d;
    edge_ids(ei, nE, e, s, d);
    float lg = edge_logit<HEADS>(as_, ad_, s, d, hh);
    float m = funkey(mkey[d * HEADS + hh]);
    float alpha = __expf(lg - m) / (ssum[d * HEADS + hh] + 1e-16f);
    const float* hrow = h + (size_t)s * (HEADS * C) + hh * C;
    float* orow = out + (size_t)d * (HEADS * C) + hh * C;
#pragma unroll
    for (int c = lane; c < C; c += 32)
        atomicAdd(orow + c, hrow[c] * alpha);
}

// p[i] = (relu?) (p[i] + b[i % width])
__global__ void bias_act(float* __restrict__ p, const float* __restrict__ b,
                         long total, int width, int relu) {
    long i = blockIdx.x * (long)blockDim.x + threadIdx.x;
    if (i >= total) return;
    float v = p[i] + b[i % width];
    p[i] = relu ? fmaxf(v, 0.f) : v;
}

extern "C" void kernel_launch(void* const* d_in, const int* in_sizes, int n_in,
                              void* d_out, int out_size, void* d_ws,
                              size_t ws_size, hipStream_t stream) {
    (void)in_sizes; (void)n_in; (void)out_size; (void)ws_size;
    const float* x    = (const float*)d_in[0];
    const int*   ei   = (const int*)d_in[1];
    const float* W1   = (const float*)d_in[2];
    const float* a_s1 = (const float*)d_in[3];
    const float* a_d1 = (const float*)d_in[4];
    const float* b1   = (const float*)d_in[5];
    const float* W2   = (const float*)d_in[6];
    const float* a_s2 = (const float*)d_in[7];
    const float* a_d2 = (const float*)d_in[8];
    const float* b2   = (const float*)d_in[9];
    float* out = (float*)d_out;

    const int N = NNODES, E = NEDGES, EE = E + N;

    // workspace layout (stream-ordered aliasing: h2 reuses h1; layer-2
    // attention scalars reuse layer-1 slots)
    float*    h1  = (float*)d_ws;                       // N*256 (then h2: N*128)
    float*    o1  = h1 + (size_t)N * 256;               // N*256
    float*    as1 = o1 + (size_t)N * 256;               // N*4
    float*    ad1 = as1 + (size_t)N * 4;                // N*4
    unsigned* m1  = (unsigned*)(ad1 + (size_t)N * 4);   // N*4
    float*    s1  = (float*)m1 + (size_t)N * 4;         // N*4

    const int TPB = 256;                 // 8 waves/block
    const int WPB = TPB / 32;

    // ---------------- layer 1: IN=128 -> H=4, C=64, concat ----------------
    hipMemsetAsync(o1, 0, (size_t)N * 256 * sizeof(float), stream);
    hipMemsetAsync(m1, 0, (size_t)N * 4 * sizeof(unsigned), stream);
    hipMemsetAsync(s1, 0, (size_t)N * 4 * sizeof(float), stream);

    // 4 waves/block, each wave = 64 output cols (4 col tiles): 256 cols/block
    gemm_wmma_f32<4, 4><<<dim3(1, N / 16), 128, 0, stream>>>(x, W1, h1, N,
                                                             128, 256);
    int w1 = N * 4;  // (node, head) waves
    node_alpha<4, 64><<<(w1 + WPB - 1) / WPB, TPB, 0, stream>>>(h1, a_s1, a_d1,
                                                                as1, ad1, N);
    int t1 = EE * 4;  // (edge, head) threads
    edge_max<4><<<(t1 + TPB - 1) / TPB, TPB, 0, stream>>>(ei, E, EE, as1, ad1, m1);
    edge_sum<4><<<(t1 + TPB - 1) / TPB, TPB, 0, stream>>>(ei, E, EE, as1, ad1,
                                                          m1, s1);
    edge_scatter<4, 64><<<(t1 + WPB - 1) / WPB, TPB, 0, stream>>>(
        ei, E, EE, h1, as1, ad1, m1, s1, o1);

    bias_act<<<((long)N * 256 + TPB - 1) / TPB, TPB, 0, stream>>>(
        o1, b1, (long)N * 256, 256, 1);   // +b1, relu

    // ---------------- layer 2: 256 -> H=1, C=128, mean(=identity) --------
    float* h2 = h1;  // reuse (layer-1 h no longer needed)
    gemm_wmma_f32<4, 4><<<dim3(1, N / 16), 128, 0, stream>>>(o1, W2, h2, N,
                                                             256, 128);
    hipMemsetAsync(m1, 0, (size_t)N * sizeof(unsigned), stream);
    hipMemsetAsync(s1, 0, (size_t)N * sizeof(float), stream);
    hipMemsetAsync(out, 0, (size_t)N * 128 * sizeof(float), stream);

    node_alpha<1, 128><<<(N + WPB - 1) / WPB, TPB, 0, stream>>>(h2, a_s2, a_d2,
                                                                as1, ad1, N);
    edge_max<1><<<(EE + TPB - 1) / TPB, TPB, 0, stream>>>(ei, E, EE, as1, ad1, m1);
    edge_sum<1><<<(EE + TPB - 1) / TPB, TPB, 0, stream>>>(ei, E, EE, as1, ad1,
                                                          m1, s1);
    edge_scatter<1, 128><<<(EE + WPB - 1) / WPB, TPB, 0, stream>>>(
        ei, E, EE, h2, as1, ad1, m1, s1, out);

    bias_act<<<((long)N * 128 + TPB - 1) / TPB, TPB, 0, stream>>>(
        out, b2, (long)N * 128, 128, 0);  // +b2
}